// AntisymmetricRNN_5617817223338
// MI455X (gfx1250) — compile-verified
//
#include <hip/hip_runtime.h>

typedef float v2f __attribute__((ext_vector_type(2)));
typedef float v4f __attribute__((ext_vector_type(4)));
typedef float v8f __attribute__((ext_vector_type(8)));

// Problem constants
#define NC    256      // state dimension n
#define TMAXC 1000     // time steps
#define COLS  16       // batch columns per workgroup (= WMMA N)
#define AS    260      // padded LDS row stride for A (floats) -> bank-conflict-free
#define XS    260      // padded LDS column stride for X (floats)

// LDS float offsets
#define A_OFF  0
#define X_OFF  (NC * AS)            // 66560
#define BY_OFF (X_OFF + COLS * XS)  // 70720
#define LDS_FLOATS (BY_OFF + NC)    // 70976  -> 283,904 bytes (< 320 KB/WGP)

__device__ __forceinline__ float hw_tanh(float x) {
#if __has_builtin(__builtin_amdgcn_tanhf)
  return __builtin_amdgcn_tanhf(x);
#elif __has_builtin(__builtin_amdgcn_tanh_f32)
  return __builtin_amdgcn_tanh_f32(x);
#else
  float y;
  // v_tanh_f32 is a TRANS op: keep a v_nop after it so the result isn't
  // consumed in the hazard window (compiler can't see inside the asm).
  asm("v_tanh_f32 %0, %1\n\tv_nop" : "=v"(y) : "v"(x));
  return y;
#endif
}

__global__ __launch_bounds__(256, 1)
void antisym_rnn_kernel(const float* __restrict__ X0,
                        const float* __restrict__ W,
                        const float* __restrict__ by,
                        float* __restrict__ out) {
  extern __shared__ float lds[];
  float* Al  = lds + A_OFF;   // A[m][k], stride AS
  float* Xl  = lds + X_OFF;   // X stored column-major: Xl[c*XS + k] = X[k][bBase+c]
  float* byl = lds + BY_OFF;

  const int tid   = threadIdx.x;
  const int bBase = blockIdx.x * COLS;

  // ---- Build A = triu(W,1) - triu(W,1)^T - eps*I into LDS ----
  for (int idx = tid; idx < NC * NC; idx += 256) {
    const int m = idx >> 8;
    const int k = idx & (NC - 1);
    float a = ((k > m) ? W[m * NC + k] : 0.0f) - ((m > k) ? W[k * NC + m] : 0.0f);
    if (m == k) a -= 0.001f;
    Al[m * AS + k] = a;
  }
  if (tid < NC) byl[tid] = by[tid];

  // ---- Init state columns from X0 (Xi0 = X0^T) and write t=0 output ----
  for (int idx = tid; idx < COLS * NC; idx += 256) {
    const int c = idx >> 8;         // column within tile
    const int k = idx & (NC - 1);   // state row
    const float x = X0[(size_t)(bBase + c) * NC + k];
    Xl[c * XS + k] = x;
    out[(size_t)(bBase + c) * TMAXC * NC + k] = x;  // out[b][0][k] = X0[b][k]
  }
  __syncthreads();

  const int lane = tid & 31;
  const int wid  = tid >> 5;     // 0..7 waves
  const int ln   = lane & 15;    // N index (batch column within tile)
  const int hi   = lane >> 4;    // half-wave selector

  // Two 16-row M tiles per wave
  const int tm0 = wid * 2;
  const int tm1 = wid * 2 + 1;
  const int m0  = tm0 * 16 + hi * 8;   // D rows this lane holds: m0..m0+7
  const int m1  = tm1 * 16 + hi * 8;

  // C init from by (broadcast over batch columns)
  v8f cb0, cb1;
#pragma unroll
  for (int r = 0; r < 8; ++r) { cb0[r] = byl[m0 + r]; cb1[r] = byl[m1 + r]; }

  // Fragment base pointers (ISA 16x16x4 f32 layouts):
  //  A 16x4: lane(ln) holds row M=ln; VGPR0/1 = K {0,1} (lanes 0-15) / {2,3} (lanes 16-31)
  //  B 4x16: lane(ln) holds col N=ln; VGPR0/1 = K {0,1} / {2,3} by half-wave
  const float* ap0  = Al + (tm0 * 16 + ln) * AS + 2 * hi;
  const float* ap1  = Al + (tm1 * 16 + ln) * AS + 2 * hi;
  const float* bp   = Xl + ln * XS + 2 * hi;
  float*       xcol = Xl + ln * XS;    // this lane's owned column for reads/updates

  for (int t = 1; t < TMAXC; ++t) {
    v8f d0 = cb0;
    v8f d1 = cb1;
#pragma unroll 8
    for (int kk = 0; kk < 64; ++kk) {
      const v2f bf = *(const v2f*)(bp  + 4 * kk);   // shared B frag for both tiles
      const v2f a0 = *(const v2f*)(ap0 + 4 * kk);
      const v2f a1 = *(const v2f*)(ap1 + 4 * kk);
      d0 = __builtin_amdgcn_wmma_f32_16x16x4_f32(false, a0, false, bf, (short)0, d0, false, false);
      d1 = __builtin_amdgcn_wmma_f32_16x16x4_f32(false, a1, false, bf, (short)0, d1, false, false);
    }

    // X_next = X + 0.01 * tanh(A X + by); old X read from LDS (still phase-1 reads)
    v8f x0n, x1n;
#pragma unroll
    for (int r = 0; r < 8; ++r) {
      x0n[r] = xcol[m0 + r] + 0.01f * hw_tanh(d0[r]);
      x1n[r] = xcol[m1 + r] + 0.01f * hw_tanh(d1[r]);
    }

    __syncthreads();   // everyone finished reading X_t

    // Update LDS state (16B-aligned: ln*XS*4 = ln*1040 ≡ 0 mod 16, m* multiple of 8)
    *(v4f*)(xcol + m0)     = __builtin_shufflevector(x0n, x0n, 0, 1, 2, 3);
    *(v4f*)(xcol + m0 + 4) = __builtin_shufflevector(x0n, x0n, 4, 5, 6, 7);
    *(v4f*)(xcol + m1)     = __builtin_shufflevector(x1n, x1n, 0, 1, 2, 3);
    *(v4f*)(xcol + m1 + 4) = __builtin_shufflevector(x1n, x1n, 4, 5, 6, 7);

    // Stream to output: out[b][t][m], 32B contiguous per lane, 32B aligned
    float* op0 = out + ((size_t)(bBase + ln) * TMAXC + t) * NC + m0;
    float* op1 = out + ((size_t)(bBase + ln) * TMAXC + t) * NC + m1;
    *(v4f*)op0       = __builtin_shufflevector(x0n, x0n, 0, 1, 2, 3);
    *(v4f*)(op0 + 4) = __builtin_shufflevector(x0n, x0n, 4, 5, 6, 7);
    *(v4f*)op1       = __builtin_shufflevector(x1n, x1n, 0, 1, 2, 3);
    *(v4f*)(op1 + 4) = __builtin_shufflevector(x1n, x1n, 4, 5, 6, 7);

    __syncthreads();   // X_{t+1} visible before next step's reads
  }
}

extern "C" void kernel_launch(void* const* d_in, const int* in_sizes, int n_in,
                              void* d_out, int out_size, void* d_ws, size_t ws_size,
                              hipStream_t stream) {
  (void)in_sizes; (void)n_in; (void)d_ws; (void)ws_size; (void)out_size;
  const float* X0 = (const float*)d_in[0];
  const float* W  = (const float*)d_in[1];
  const float* by = (const float*)d_in[2];
  float* out = (float*)d_out;

  const int ldsBytes = LDS_FLOATS * 4;  // 283,904 B (< 320 KB per WGP on CDNA5)
  (void)hipFuncSetAttribute((const void*)antisym_rnn_kernel,
                            hipFuncAttributeMaxDynamicSharedMemorySize, ldsBytes);

  antisym_rnn_kernel<<<dim3(256 / COLS), dim3(256), ldsBytes, stream>>>(X0, W, by, out);
}